// RegionExtractor_86792699117980
// MI455X (gfx1250) — compile-verified
//
#include <hip/hip_runtime.h>
#include <hip/hip_bf16.h>

#define BATCH   16
#define NBOX    4096
#define MAXDET  100
#define NT      1024
#define CH      16            // greedy chunk width == WMMA tile dim
#define MIN_SZ  25.0f
#define IOU_T   0.3f

typedef __attribute__((ext_vector_type(2))) float v2f;
typedef __attribute__((ext_vector_type(8))) float v8f;

__global__ __launch_bounds__(NT) void RegionExtractor_nms_kernel(
    const float* __restrict__ boxes,     // [B, N, 4] xyxy
    const float* __restrict__ scores,    // [B, N]
    float* __restrict__ out)             // [B, MAXDET, 5]
{
    // Per-image working set lives entirely in LDS (CDNA5: 320 KB / WGP).
    __shared__ float4             raw[NBOX];            // 64 KB, async-DMA staged
    __shared__ unsigned long long keys[NBOX];           // 32 KB sort keys
    __shared__ unsigned           keep[NBOX / 32];      // 512 B keep bitmask
    __shared__ unsigned           wpre[NBOX / 32 + 1];  // word prefix sums

    const int img  = blockIdx.x;
    const int tid  = threadIdx.x;
    const int wid  = tid >> 5;
    const int lane = tid & 31;
    const float4* __restrict__ bsrc = (const float4*)(boxes + (size_t)img * NBOX * 4);

    // ---------------- Phase 0: async global -> LDS DMA of the box tile ----
    {
        unsigned lds_base = (unsigned)(size_t)(&raw[0]);
        #pragma unroll
        for (int it = 0; it < NBOX / NT; ++it) {
            int i = it * NT + tid;
            unsigned dst = lds_base + (unsigned)i * 16u;
            const float4* src = bsrc + i;
            asm volatile("global_load_async_to_lds_b128 %0, %1, off"
                         :: "v"(dst), "v"(src) : "memory");
        }
        asm volatile("s_wait_asynccnt 0" ::: "memory");
    }
    __syncthreads();

    // ---------------- Phase 1: build sort keys ----------------------------
    // ascending sort on key == (score descending, index ascending)
    #pragma unroll
    for (int it = 0; it < NBOX / NT; ++it) {
        int i = it * NT + tid;
        float4 b = raw[i];
        bool valid = ((b.z - b.x) >= MIN_SZ) && ((b.w - b.y) >= MIN_SZ);
        float s = valid ? scores[(size_t)img * NBOX + i] : -__builtin_inff();
        unsigned u = __float_as_uint(s);
        u = (u & 0x80000000u) ? ~u : (u | 0x80000000u);  // total-order map (asc)
        keys[i] = ((unsigned long long)(~u) << 32) | (unsigned)i;
    }
    __syncthreads();

    // ---------------- Phase 2: bitonic sort (ascending) in LDS ------------
    for (unsigned k = 2; k <= NBOX; k <<= 1) {
        for (unsigned j = k >> 1; j > 0; j >>= 1) {
            for (unsigned t = (unsigned)tid; t < NBOX; t += NT) {
                unsigned ixj = t ^ j;
                if (ixj > t) {
                    unsigned long long a = keys[t];
                    unsigned long long b = keys[ixj];
                    bool up = ((t & k) == 0);
                    if ((a > b) == up) { keys[t] = b; keys[ixj] = a; }
                }
            }
            __syncthreads();
        }
    }

    // ---------------- Phase 3: init keep bitmask (wave32 ballot) ----------
    #pragma unroll
    for (int it = 0; it < NBOX / NT; ++it) {
        int p = it * NT + tid;
        unsigned idx = (unsigned)keys[p] & (NBOX - 1);
        float4 b = raw[idx];
        bool valid = ((b.z - b.x) >= MIN_SZ) && ((b.w - b.y) >= MIN_SZ);
        unsigned long long m = __ballot(valid);     // wave32: low 32 bits
        if ((tid & 31) == 0) keep[p >> 5] = (unsigned)m;
    }
    __syncthreads();

    // ---------------- Phase 4: chunked greedy NMS -------------------------
    // 16-rank chunks: serial resolution in registers (blockwide-identical),
    // then bulk suppression of later ranks in 16x16 tiles, one wave each:
    // area_m + barea_n tile via V_WMMA_F32_16X16X4_F32, intersection on VALU.
    for (int c = 0; c < NBOX; c += CH) {
        const unsigned shift = (unsigned)(c & 31);
        const unsigned orig  = (keep[c >> 5] >> shift) & 0xFFFFu;
        unsigned sub = orig;

        // ---- intra-chunk greedy (redundant per thread, identical result)
        for (int a = 0; a < CH; ++a) {
            if (!((sub >> a) & 1u)) continue;
            unsigned ka = (unsigned)keys[c + a] & (NBOX - 1);
            float4 A = raw[ka];
            float aarea = (A.z - A.x) * (A.w - A.y);
            for (int l = a + 1; l < CH; ++l) {
                if (!((sub >> l) & 1u)) continue;
                unsigned kl = (unsigned)keys[c + l] & (NBOX - 1);
                float4 L = raw[kl];
                float ix = fmaxf(fminf(A.z, L.z) - fmaxf(A.x, L.x), 0.0f);
                float iy = fmaxf(fminf(A.w, L.w) - fmaxf(A.y, L.y), 0.0f);
                float inter = ix * iy;
                float larea = (L.z - L.x) * (L.w - L.y);
                float iou = inter / (aarea + larea - inter + 1e-9f);
                if (iou > IOU_T) sub &= ~(1u << l);
            }
        }
        // write back intra-chunk suppressions (only read again in phase 5)
        if (tid == 0) {
            unsigned cleared = orig & ~sub;
            if (cleared) atomicAnd(&keep[c >> 5], ~(cleared << shift));
        }
        if (sub == 0u || c + CH >= NBOX) continue;   // nothing to apply

        // ---- apply chunk's kept boxes to all later ranks, 16-j tiles/wave
        const int ntiles = (NBOX - c - CH) >> 4;
        for (int t = wid; t < ntiles; t += NT / 32) {
            const int tj = c + CH + (t << 4);
            const int n  = lane & 15;
            const int j  = tj + n;

            unsigned jw  = keep[j >> 5];
            bool jalive  = (jw >> (j & 31)) & 1u;
            if (__ballot(jalive) == 0ull) continue;  // wave-uniform skip (pre-WMMA)

            unsigned kj = (unsigned)keys[j] & (NBOX - 1);
            float4 Bb = raw[kj];
            float barea = (Bb.z - Bb.x) * (Bb.w - Bb.y);

            unsigned km = (unsigned)keys[c + n] & (NBOX - 1);
            float4 Am = raw[km];
            float marea = (Am.z - Am.x) * (Am.w - Am.y);

            // A row m = (area_m, 1, 1, 0); B col n = (1, barea_n, 0, 0):
            // D(m,n) = area_m + barea_n under either K lane-half ordering of B.
            const bool hi = lane >= 16;               // lanes 16-31: K=2,3 slots
            v2f A2, B2;
            A2.x = hi ? 1.0f : marea;
            A2.y = hi ? 0.0f : 1.0f;
            B2.x = hi ? 0.0f : 1.0f;
            B2.y = hi ? 0.0f : barea;
            v8f S = {};
            S = __builtin_amdgcn_wmma_f32_16x16x4_f32(
                    false, A2, false, B2, (short)0, S, false, false);

            // C/D layout: lanes 0-15 VGPRk=(M=k,N=lane); lanes 16-31 M=k+8.
            unsigned sup = 0u;
            const int mbase = hi ? 8 : 0;
            if (jalive) {
                #pragma unroll
                for (int kk = 0; kk < 8; ++kk) {
                    int m = mbase + kk;
                    if (!((sub >> m) & 1u)) continue;
                    unsigned ka = (unsigned)keys[c + m] & (NBOX - 1);
                    float4 Ab = raw[ka];
                    float ix = fmaxf(fminf(Ab.z, Bb.z) - fmaxf(Ab.x, Bb.x), 0.0f);
                    float iy = fmaxf(fminf(Ab.w, Bb.w) - fmaxf(Ab.y, Bb.y), 0.0f);
                    float inter = ix * iy;
                    // iou > T  <=>  inter > T * (areasum - inter + eps)
                    if (inter > IOU_T * (S[kk] - inter + 1e-9f)) sup = 1u;
                }
            }
            sup |= (unsigned)__shfl_xor((int)sup, 16, 32);  // merge M halves
            if (lane < 16 && sup)
                atomicAnd(&keep[j >> 5], ~(1u << (j & 31)));
        }
        __syncthreads();   // publish suppressions before next chunk resolves
    }
    __syncthreads();

    // ---------------- Phase 5: rank kept boxes, emit first MAXDET ---------
    if (tid == 0) {
        unsigned acc = 0;
        for (int w = 0; w < NBOX / 32; ++w) { wpre[w] = acc; acc += __popc(keep[w]); }
        wpre[NBOX / 32] = acc;
    }
    __syncthreads();

    const unsigned total = wpre[NBOX / 32];
    float* __restrict__ obase = out + (size_t)img * MAXDET * 5;

    #pragma unroll
    for (int it = 0; it < NBOX / NT; ++it) {
        int p = it * NT + tid;
        unsigned w = keep[p >> 5];
        if ((w >> (p & 31)) & 1u) {
            unsigned rank = wpre[p >> 5] + __popc(w & ((1u << (p & 31)) - 1u));
            if (rank < MAXDET) {
                unsigned idx = (unsigned)keys[p] & (NBOX - 1);
                float4 b = raw[idx];
                float* r = obase + rank * 5;
                r[0] = (float)img;
                r[1] = b.x; r[2] = b.y; r[3] = b.z; r[4] = b.w;
            }
        }
    }
    // pad rows: (-1, 0,0,0,0), same as reference's fixed-shape padding
    for (int r = tid; r < MAXDET; r += NT) {
        if ((unsigned)r >= total) {
            float* o = obase + r * 5;
            o[0] = -1.0f; o[1] = 0.0f; o[2] = 0.0f; o[3] = 0.0f; o[4] = 0.0f;
        }
    }
}

extern "C" void kernel_launch(void* const* d_in, const int* in_sizes, int n_in,
                              void* d_out, int out_size, void* d_ws, size_t ws_size,
                              hipStream_t stream) {
    const float* boxes  = (const float*)d_in[0];   // [16, 4096, 4] f32
    const float* scores = (const float*)d_in[1];   // [16, 4096]    f32
    float* out = (float*)d_out;                    // [16, 100, 5]  f32
    (void)in_sizes; (void)n_in; (void)out_size; (void)d_ws; (void)ws_size;
    RegionExtractor_nms_kernel<<<BATCH, NT, 0, stream>>>(boxes, scores, out);
}